// CustomConvolutionLayer_12618613916213
// MI455X (gfx1250) — compile-verified
//
#include <hip/hip_runtime.h>

typedef __attribute__((ext_vector_type(2))) float v2f;
typedef __attribute__((ext_vector_type(8))) float v8f;

#define INPUT_DIM 96
#define KDIM 5
#define ODIM 92                       // 96 - 5 + 1
#define OSIZE (ODIM * ODIM)           // 8464
#define ISIZE (INPUT_DIM * INPUT_DIM) // 9216
#define BATCH 64
#define NTILE_B 4                     // 64 / 16 batch tiles (one per wave)
#define NTILE_OJ 6                    // ceil(92/16) output-column tiles

// Grid: (oi = 92, ojt = 6), block = 128 threads = 4 waves, wave id = batch tile.
// One wave32 computes a 16(batch) x 16(oj) f32 output tile for one output row
// via 25x V_WMMA_F32_16X16X4_F32 (5 kernel rows x 5 K=4 chunks over the
// 20-wide Toeplitz band of input columns).
__global__ __launch_bounds__(128)
void conv_wmma_f32(const float* __restrict__ x,
                   const float* __restrict__ kern,
                   float* __restrict__ out) {
    const int lane  = threadIdx.x & 31;
    const int btile = threadIdx.x >> 5;        // 0..3  (batch tile = wave id)
    const int oi    = blockIdx.x;              // 0..91 (output row)
    const int oj0   = blockIdx.y << 4;         // 0..80 (output col base == ij0)
    const int b0    = btile << 4;

    const int n     = lane & 15;   // A: batch row m == n ; B/C/D: column index
    const int khalf = lane >> 4;   // lane half selects K pair {0,1} vs {2,3}

    // Kernel taps: uniform address -> scalar loads, live in SGPRs.
    float k5[KDIM][KDIM];
#pragma unroll
    for (int i = 0; i < KDIM; ++i)
#pragma unroll
        for (int j = 0; j < KDIM; ++j)
            k5[i][j] = kern[i * KDIM + j];

    v8f c = {};   // f32 accumulator, 8 VGPRs

    const float* xb = x + (size_t)(b0 + n) * ISIZE;   // this lane's batch row

#pragma unroll
    for (int ki = 0; ki < KDIM; ++ki) {
        const int ii = oi + ki;                       // input row, <= 95
        const float* xrow = xb + ii * INPUT_DIM;
#pragma unroll
        for (int ch = 0; ch < 5; ++ch) {
            // ---- A fragment: x[b0+n, ii, oj0 + 4*ch + 2*khalf + {0,1}] ----
            int ij  = oj0 + 4 * ch + 2 * khalf;       // even -> 8B aligned
            int ijc = ij > (INPUT_DIM - 2) ? (INPUT_DIM - 2) : ij; // clamp (stays even)
            v2f a = *(const v2f*)(xrow + ijc);
            // Clamped lanes only ever multiply structurally-zero B entries
            // for valid output columns, so no zeroing needed.

            // ---- B fragment: B[k, n] = kern[ki][4*ch + k - n] (Toeplitz band) ----
            const int d0 = 4 * ch + 2 * khalf - n;    // kj for K = 2*khalf
            const int d1 = d0 + 1;                    // kj for K = 2*khalf+1
            float b_lo = 0.0f, b_hi = 0.0f;
#pragma unroll
            for (int t = 0; t < KDIM; ++t) {
                b_lo = (d0 == t) ? k5[ki][t] : b_lo;
                b_hi = (d1 == t) ? k5[ki][t] : b_hi;
            }
            v2f b;
            b.x = b_lo;
            b.y = b_hi;

            // D = A x B + C  (exact f32 math, same as reference)
            c = __builtin_amdgcn_wmma_f32_16x16x4_f32(
                    /*neg_a=*/false, a, /*neg_b=*/false, b,
                    /*c_mod=*/(short)0, c, /*reuse_a=*/false, /*reuse_b=*/false);
        }
    }

    // ---- Store: VGPR v holds rows M = v + 8*khalf, column N = n ----
    const int oj = oj0 + n;
    if (oj < ODIM) {                               // divergent only in last ojt
        const size_t pcol = (size_t)oi * ODIM + oj;
#pragma unroll
        for (int v = 0; v < 8; ++v) {
            const int mb = v + 8 * khalf;
            out[(size_t)(b0 + mb) * OSIZE + pcol] = c[v];
        }
    }
}

extern "C" void kernel_launch(void* const* d_in, const int* in_sizes, int n_in,
                              void* d_out, int out_size, void* d_ws, size_t ws_size,
                              hipStream_t stream) {
    const float* x    = (const float*)d_in[0];   // [64, 9216] f32
    const float* kern = (const float*)d_in[1];   // [5, 5] f32
    float* out        = (float*)d_out;           // [64, 8464] f32

    dim3 grid(ODIM, NTILE_OJ);                   // (92, 6)
    conv_wmma_f32<<<grid, 128, 0, stream>>>(x, kern, out);
}